// ResidueIntraBlockGNN_59931973649025
// MI455X (gfx1250) — compile-verified
//
#include <hip/hip_runtime.h>

#define N_NODES 16384
#define N_EDGES 262144
#define DIM     256

typedef __attribute__((ext_vector_type(2))) float v2f;
typedef __attribute__((ext_vector_type(8))) float v8f;

// ---------------------------------------------------------------------------
// 1) deg init: every node has a self-loop of weight 1
// ---------------------------------------------------------------------------
__global__ __launch_bounds__(256) void init_deg_kernel(float* __restrict__ deg) {
    int i = blockIdx.x * 256 + threadIdx.x;
    if (i < N_NODES) deg[i] = 1.0f;
}

// ---------------------------------------------------------------------------
// 2) deg accumulate over intra-block edges (mask = sec_ids match)
// ---------------------------------------------------------------------------
__global__ __launch_bounds__(256) void edge_deg_kernel(const int* __restrict__ ei,
                                                       const int* __restrict__ sec,
                                                       float* __restrict__ deg) {
    int e = blockIdx.x * 256 + threadIdx.x;
    if (e >= N_EDGES) return;
    int row = ei[e];            // edge_index[0][e] (message source)
    int col = ei[N_EDGES + e];  // edge_index[1][e] (message target)
    if (sec[row] == sec[col]) {
        unsafeAtomicAdd(&deg[col], 1.0f);   // -> global_atomic_add_f32
    }
}

// ---------------------------------------------------------------------------
// 3) dinv = rsqrt(deg)  (deg >= 1 always, self-loop guarantees it)
// ---------------------------------------------------------------------------
__global__ __launch_bounds__(256) void dinv_kernel(float* __restrict__ deg) {
    int i = blockIdx.x * 256 + threadIdx.x;
    if (i < N_NODES) deg[i] = rsqrtf(deg[i]);
}

// ---------------------------------------------------------------------------
// 4) h = x @ W via V_WMMA_F32_16X16X4_F32.
//    Block = 256 threads = 8 waves; block tile = 128 rows x 64 cols of h.
//    Each wave: 16 rows x 64 cols = 4 independent accumulators sharing one A
//    operand per K-step -> 4 parallel WMMA dependency chains (hides XDL
//    latency) and 4x fewer global reads of x (64 MB instead of 256 MB L2
//    traffic across the grid).
//
//    W column panel (256 x 64) staged in LDS as 4 conflict-free 256x16
//    panels, 64 KB total (WGP budget is 320 KB).
//
//    fp32 WMMA operand layout (ISA 7.12.2):
//      A (16x4, 2 VGPRs): lane L -> M = L%16 ; VGPRj holds K = 2*(L/16) + j
//      B (4x16, 2 VGPRs): lane L -> N = L%16 ; VGPRj holds K = 2*(L/16) + j
//      C/D (16x16, 8 VGPRs): VGPRv, lane L -> M = v + 8*(L/16), N = L%16
// ---------------------------------------------------------------------------
__global__ __launch_bounds__(256) void gemm_wmma_kernel(const float* __restrict__ x,
                                                        const float* __restrict__ W,
                                                        float* __restrict__ h) {
    __shared__ float wt[4][DIM * 16];       // 4 panels, [k][n] layout each

    const int n0 = blockIdx.x * 64;                             // column tile
    const int m0 = blockIdx.y * 128 + (threadIdx.x >> 5) * 16;  // per-wave rows

    // cooperative stage of W[:, n0:n0+64] into LDS (thread t loads row k = t)
    {
        const int k = threadIdx.x;  // 0..255 == DIM rows
        const float4* src = reinterpret_cast<const float4*>(W + k * DIM + n0);
#pragma unroll
        for (int t = 0; t < 4; ++t) {
            float4* dst = reinterpret_cast<float4*>(&wt[t][k * 16]);
            dst[0] = src[t * 4 + 0];
            dst[1] = src[t * 4 + 1];
            dst[2] = src[t * 4 + 2];
            dst[3] = src[t * 4 + 3];
        }
    }
    __syncthreads();

    const int lane  = threadIdx.x & 31;
    const int mrow  = lane & 15;            // A: M index
    const int ncol  = lane & 15;            // B/C: N index
    const int khalf = (lane >> 4) << 1;     // 0 or 2: K sub-offset per half-wave

    const float* xrow = x + (size_t)(m0 + mrow) * DIM;

    v8f c0 = {}, c1 = {}, c2 = {}, c3 = {};
#pragma unroll 4
    for (int k0 = 0; k0 < DIM; k0 += 4) {
        v2f a;
        a.x = xrow[k0 + khalf];              // global_load_b64 (consecutive)
        a.y = xrow[k0 + khalf + 1];

        v2f b0, b1, b2, b3;
        b0.x = wt[0][(k0 + khalf) * 16 + ncol];
        b0.y = wt[0][(k0 + khalf + 1) * 16 + ncol];
        b1.x = wt[1][(k0 + khalf) * 16 + ncol];
        b1.y = wt[1][(k0 + khalf + 1) * 16 + ncol];
        b2.x = wt[2][(k0 + khalf) * 16 + ncol];
        b2.y = wt[2][(k0 + khalf + 1) * 16 + ncol];
        b3.x = wt[3][(k0 + khalf) * 16 + ncol];
        b3.y = wt[3][(k0 + khalf + 1) * 16 + ncol];

        // 4 independent accumulator chains sharing one A operand
        c0 = __builtin_amdgcn_wmma_f32_16x16x4_f32(false, a, false, b0, (short)0, c0, false, false);
        c1 = __builtin_amdgcn_wmma_f32_16x16x4_f32(false, a, false, b1, (short)0, c1, false, false);
        c2 = __builtin_amdgcn_wmma_f32_16x16x4_f32(false, a, false, b2, (short)0, c2, false, false);
        c3 = __builtin_amdgcn_wmma_f32_16x16x4_f32(false, a, false, b3, (short)0, c3, false, false);
    }

    const int mhi = (lane >> 4) * 8;
    float* hbase = h + (size_t)(m0 + mhi) * DIM + n0 + ncol;
#pragma unroll
    for (int v = 0; v < 8; ++v) {
        hbase[(size_t)v * DIM + 0]  = c0[v];
        hbase[(size_t)v * DIM + 16] = c1[v];
        hbase[(size_t)v * DIM + 32] = c2[v];
        hbase[(size_t)v * DIM + 48] = c3[v];
    }
}

// ---------------------------------------------------------------------------
// 5) out = b + dinv[i]^2 * h[i]   (self-loop contribution + bias), float4
// ---------------------------------------------------------------------------
__global__ __launch_bounds__(256) void self_bias_kernel(const float4* __restrict__ h4,
                                                        const float* __restrict__ dinv,
                                                        const float4* __restrict__ b4,
                                                        float4* __restrict__ out4) {
    int idx  = blockIdx.x * 256 + threadIdx.x;  // float4 index, DIM/4 = 64 per node
    int node = idx >> 6;
    int f4   = idx & 63;
    float s  = dinv[node];
    s = s * s;
    float4 hv = h4[idx];
    float4 bv = b4[f4];
    float4 o;
    o.x = bv.x + s * hv.x;
    o.y = bv.y + s * hv.y;
    o.z = bv.z + s * hv.z;
    o.w = bv.w + s * hv.w;
    out4[idx] = o;
}

// ---------------------------------------------------------------------------
// 6) masked edge scatter: one wave per edge; ~1/200 edges survive the mask,
//    so the wave-uniform compare lets 99.5% of waves exit via s_cbranch.
//    Each lane handles 8 features (2x float4 gather, 8 fp32 atomics).
// ---------------------------------------------------------------------------
__global__ __launch_bounds__(256) void edge_scatter_kernel(const int* __restrict__ ei,
                                                           const int* __restrict__ sec,
                                                           const float* __restrict__ dinv,
                                                           const float* __restrict__ h,
                                                           float* __restrict__ out) {
    int e = blockIdx.x * 8 + (threadIdx.x >> 5);   // wave-uniform edge id
    if (e >= N_EDGES) return;
    int row = ei[e];
    int col = ei[N_EDGES + e];
    if (sec[row] != sec[col]) return;              // drop inter-block edges

    float s = dinv[row] * dinv[col];
    int lane = threadIdx.x & 31;

    const float4* hs = reinterpret_cast<const float4*>(h + (size_t)row * DIM + lane * 8);
    float4 h0 = hs[0];
    float4 h1 = hs[1];
    float* op = out + (size_t)col * DIM + lane * 8;
    unsafeAtomicAdd(op + 0, s * h0.x);
    unsafeAtomicAdd(op + 1, s * h0.y);
    unsafeAtomicAdd(op + 2, s * h0.z);
    unsafeAtomicAdd(op + 3, s * h0.w);
    unsafeAtomicAdd(op + 4, s * h1.x);
    unsafeAtomicAdd(op + 5, s * h1.y);
    unsafeAtomicAdd(op + 6, s * h1.z);
    unsafeAtomicAdd(op + 7, s * h1.w);
}

// ---------------------------------------------------------------------------
extern "C" void kernel_launch(void* const* d_in, const int* in_sizes, int n_in,
                              void* d_out, int out_size, void* d_ws, size_t ws_size,
                              hipStream_t stream) {
    const float* x   = (const float*)d_in[0];   // [16384, 256]
    const float* W   = (const float*)d_in[1];   // [256, 256]
    const float* b   = (const float*)d_in[2];   // [256]
    const int*   ei  = (const int*)d_in[3];     // [2, 262144]
    const int*   sec = (const int*)d_in[4];     // [16384]
    float*       out = (float*)d_out;           // [16384, 256]

    // workspace: dinv (64 KB, keeps h float4-aligned) then h (16 MB)
    float* dinv = (float*)d_ws;
    float* h    = dinv + N_NODES;

    // 1-3: symmetric normalization coefficients
    init_deg_kernel<<<N_NODES / 256, 256, 0, stream>>>(dinv);
    edge_deg_kernel<<<N_EDGES / 256, 256, 0, stream>>>(ei, sec, dinv);
    dinv_kernel<<<N_NODES / 256, 256, 0, stream>>>(dinv);

    // 4: h = x @ W  (grid: 4 column tiles x 128 row groups; 128x64 per block)
    gemm_wmma_kernel<<<dim3(DIM / 64, N_NODES / 128), 256, 0, stream>>>(x, W, h);

    // 5: out = b + dinv^2 * h (self loops)
    self_bias_kernel<<<(N_NODES * DIM / 4) / 256, 256, 0, stream>>>(
        (const float4*)h, dinv, (const float4*)b, (float4*)out);

    // 6: scatter masked edge messages
    edge_scatter_kernel<<<N_EDGES / 8, 256, 0, stream>>>(ei, sec, dinv, h, out);
}